// GraphConvolutionalLayers_34711925686444
// MI455X (gfx1250) — compile-verified
//
#include <hip/hip_runtime.h>
#include <hip/hip_bf16.h>

#define HID_F 128
#define EPS_LN 1e-5f

typedef float v2f __attribute__((ext_vector_type(2)));
typedef float v8f __attribute__((ext_vector_type(8)));

// ---------------- degree / dinv ----------------
__global__ void k_deg_init(float* deg, int N) {
  int i = blockIdx.x * blockDim.x + threadIdx.x;
  if (i < N) deg[i] = 1.0f;   // self loop
}

__global__ void k_deg_count(const int* __restrict__ dst, float* deg, int E) {
  int e = blockIdx.x * blockDim.x + threadIdx.x;
  if (e < E) unsafeAtomicAdd(&deg[dst[e]], 1.0f);
}

__global__ void k_deg_rsqrt(float* deg, int N) {
  int i = blockIdx.x * blockDim.x + threadIdx.x;
  if (i < N) deg[i] = rsqrtf(deg[i]);   // deg buffer becomes dinv
}

// ---------------- GEMM: hw = H @ W ; agg = hw * dinv^2 (self loop) ----------------
// One wave computes one 16x16 tile with V_WMMA_F32_16X16X4_F32, K=128 in 32 steps.
// Block = 256 threads = 8 waves -> 16 rows x 128 cols per block.
__global__ __launch_bounds__(256) void k_gemm_wmma(
    const float* __restrict__ H, const float* __restrict__ W,
    const float* __restrict__ dinv, float* __restrict__ HWout,
    float* __restrict__ Agg, int N) {
  const int lane  = threadIdx.x & 31;
  const int wave  = threadIdx.x >> 5;
  const int row0  = blockIdx.x << 4;     // M tile
  const int col0  = wave << 4;           // N tile (8 waves cover 128 cols)
  const int l15   = lane & 15;
  const int khalf = (lane >> 4) << 1;    // lanes 16..31 handle K+2,K+3

  // clamp row for address safety (N is a multiple of 16 in practice)
  const int rowA  = min(row0 + l15, N - 1);
  const float* arow = H + (size_t)rowA * HID_F + khalf;
  const float* bcol = W + (size_t)khalf * HID_F + col0 + l15;

  v8f acc = {};
#pragma unroll 8
  for (int k = 0; k < HID_F; k += 4) {
    v2f a, b;
    // A fragment: lane l (0-15): A[M=l][k], A[M=l][k+1]; lanes 16-31: K+2,K+3
    a.x = arow[k];
    a.y = arow[k + 1];
    // B fragment: lane l (0-15): B[k][col0+l], B[k+1][col0+l]; lanes 16-31: K+2,K+3
    b.x = bcol[(size_t)k * HID_F];
    b.y = bcol[(size_t)(k + 1) * HID_F];
    acc = __builtin_amdgcn_wmma_f32_16x16x4_f32(
        /*neg_a=*/false, a, /*neg_b=*/false, b,
        /*c_mod=*/(short)0, acc, /*reuse_a=*/false, /*reuse_b=*/false);
  }

  // C/D layout: VGPR r -> lanes 0-15: M=r, lanes 16-31: M=r+8; N = lane&15
  const int colg  = col0 + l15;
  const int mbase = row0 + ((lane >> 4) << 3);
#pragma unroll
  for (int r = 0; r < 8; ++r) {
    int m = mbase + r;
    if (m < N) {
      float v  = acc[r];
      float di = dinv[m];
      HWout[(size_t)m * HID_F + colg] = v;
      Agg[(size_t)m * HID_F + colg]   = v * di * di;   // self-loop term
    }
  }
}

// ---------------- edge scatter: agg[dst] += hw[src] * dinv[src]*dinv[dst] ----------------
// One wave per edge; each lane handles 4 contiguous features (float4).
__global__ __launch_bounds__(256) void k_edge_scatter(
    const float* __restrict__ hw, const float* __restrict__ dinv,
    const int* __restrict__ src, const int* __restrict__ dst,
    float* __restrict__ agg, int E) {
  int e = blockIdx.x * (blockDim.x >> 5) + (threadIdx.x >> 5);
  if (e >= E) return;
  int lane = threadIdx.x & 31;
  int s = src[e], d = dst[e];
  float w = dinv[s] * dinv[d];
  float4 v = ((const float4*)(hw + (size_t)s * HID_F))[lane];
  float* ap = agg + (size_t)d * HID_F + (lane << 2);
  unsafeAtomicAdd(ap + 0, v.x * w);
  unsafeAtomicAdd(ap + 1, v.y * w);
  unsafeAtomicAdd(ap + 2, v.z * w);
  unsafeAtomicAdd(ap + 3, v.w * w);
}

// ---------------- LayerNorm + ReLU + bias + optional residual ----------------
// One wave per node; 128 features = 32 lanes x float4; wave32 shuffle reductions.
__global__ __launch_bounds__(256) void k_ln_relu(
    const float* __restrict__ agg, const float* __restrict__ bias,
    const float* __restrict__ gamma, const float* __restrict__ beta,
    const float* __restrict__ resid, float* __restrict__ out,
    int N, int addResid) {
  int n = blockIdx.x * (blockDim.x >> 5) + (threadIdx.x >> 5);
  if (n >= N) return;
  int lane = threadIdx.x & 31;

  float4 v = ((const float4*)(agg + (size_t)n * HID_F))[lane];
  float4 b = ((const float4*)bias)[lane];
  v.x += b.x; v.y += b.y; v.z += b.z; v.w += b.w;

  float s = v.x + v.y + v.z + v.w;
#pragma unroll
  for (int o = 16; o > 0; o >>= 1) s += __shfl_xor(s, o);
  float mu = s * (1.0f / HID_F);

  float dx = v.x - mu, dy = v.y - mu, dz = v.z - mu, dw = v.w - mu;
  float q = dx * dx + dy * dy + dz * dz + dw * dw;
#pragma unroll
  for (int o = 16; o > 0; o >>= 1) q += __shfl_xor(q, o);
  float rinv = rsqrtf(q * (1.0f / HID_F) + EPS_LN);

  float4 g  = ((const float4*)gamma)[lane];
  float4 be = ((const float4*)beta)[lane];
  float4 o4;
  o4.x = fmaxf(g.x * dx * rinv + be.x, 0.0f);
  o4.y = fmaxf(g.y * dy * rinv + be.y, 0.0f);
  o4.z = fmaxf(g.z * dz * rinv + be.z, 0.0f);
  o4.w = fmaxf(g.w * dw * rinv + be.w, 0.0f);

  if (addResid) {
    float4 r = ((const float4*)(resid + (size_t)n * HID_F))[lane];
    o4.x += r.x; o4.y += r.y; o4.z += r.z; o4.w += r.w;
  }
  ((float4*)(out + (size_t)n * HID_F))[lane] = o4;
}

// ---------------- host-side orchestration ----------------
extern "C" void kernel_launch(void* const* d_in, const int* in_sizes, int n_in,
                              void* d_out, int out_size, void* d_ws, size_t ws_size,
                              hipStream_t stream) {
  const int N = in_sizes[0] / HID_F;   // 50000
  const int E = in_sizes[1];           // 640000

  const float* x   = (const float*)d_in[0];
  const int*   src = (const int*)d_in[1];
  const int*   dst = (const int*)d_in[2];
  const float* Wl[3]  = {(const float*)d_in[3], (const float*)d_in[7],  (const float*)d_in[11]};
  const float* bl[3]  = {(const float*)d_in[4], (const float*)d_in[8],  (const float*)d_in[12]};
  const float* gl[3]  = {(const float*)d_in[5], (const float*)d_in[9],  (const float*)d_in[13]};
  const float* bel[3] = {(const float*)d_in[6], (const float*)d_in[10], (const float*)d_in[14]};

  // workspace layout
  char* wsb = (char*)d_ws;
  size_t nodeVecBytes = (size_t)N * HID_F * sizeof(float);
  size_t off = 0;
  float* dinv = (float*)(wsb + off); off += ((size_t)N * sizeof(float) + 255) & ~(size_t)255;
  float* hw   = (float*)(wsb + off); off += nodeVecBytes;
  float* agg  = (float*)(wsb + off); off += nodeVecBytes;
  float* bufA = (float*)(wsb + off); off += nodeVecBytes;
  float* bufB = (float*)(wsb + off); off += nodeVecBytes;
  (void)ws_size; (void)n_in;

  // degrees -> dinv
  k_deg_init<<<(N + 255) / 256, 256, 0, stream>>>(dinv, N);
  k_deg_count<<<(E + 255) / 256, 256, 0, stream>>>(dst, dinv, E);
  k_deg_rsqrt<<<(N + 255) / 256, 256, 0, stream>>>(dinv, N);

  const int gemmBlocks = (N + 15) / 16;      // 16 rows per block
  const int edgeBlocks = (E + 7) / 8;        // 8 edges (waves) per block
  const int nodeBlocks = (N + 7) / 8;        // 8 nodes (waves) per block

  const float* h = x;                        // layer input
  float* houts[3] = {bufA, bufB, (float*)d_out};
  for (int layer = 0; layer < 3; ++layer) {
    k_gemm_wmma<<<gemmBlocks, 256, 0, stream>>>(h, Wl[layer], dinv, hw, agg, N);
    k_edge_scatter<<<edgeBlocks, 256, 0, stream>>>(hw, dinv, src, dst, agg, E);
    k_ln_relu<<<nodeBlocks, 256, 0, stream>>>(
        agg, bl[layer], gl[layer], bel[layer],
        /*resid=*/h, houts[layer], N, /*addResid=*/(layer > 0) ? 1 : 0);
    h = houts[layer];
  }
}